// DepthDCOp_40389872451640
// MI455X (gfx1250) — compile-verified
//
#include <hip/hip_runtime.h>

// DepthDC: out[n,c,h,w] = sum_{kh,kw} x_pad[n,c,h+kh-1,w+kw-1] * kern[n,0,kh*3+kw,h,w]
// N=8, C=256, H=W=64, K=3, dilation=1.  Memory-bound (~68MB moved, ~0.15 GFLOP).

#define NN 8
#define CC 256
#define HH 64
#define WW 64
#define PP 9          // K*K
#define CB 32         // channels per block
#define TPB 256       // 8 waves (wave32)

// ---- CDNA5 async global->LDS copy (ASYNCcnt-tracked) -----------------------
__device__ __forceinline__ void async_b128_to_lds(unsigned lds_byte_off,
                                                  const void* gaddr) {
    // VDST = LDS byte address (one VGPR), VADDR = 64-bit global address (VGPR pair)
    asm volatile("global_load_async_to_lds_b128 %0, %1, off"
                 :: "v"(lds_byte_off), "v"(gaddr)
                 : "memory");
}

__device__ __forceinline__ void wait_async_all() {
    asm volatile("s_wait_asynccnt 0" ::: "memory");
}
// ---------------------------------------------------------------------------

__global__ __launch_bounds__(TPB) void ddc_kernel(const float* __restrict__ x,
                                                  const float* __restrict__ kern,
                                                  float* __restrict__ out) {
    // 9 filter-tap rows of W floats, shared by all 32 channels in this block.
    __shared__ float kw_lds[PP][WW];   // 2304 B

    const int h   = blockIdx.x;        // output row (block-uniform)
    const int c0  = blockIdx.y * CB;   // channel group base
    const int n   = blockIdx.z;        // image
    const int tid = threadIdx.x;

    // ---- Stage kernel taps into LDS asynchronously: 9*64 floats = 144 x b128
    if (tid < PP * (WW / 4)) {
        const int p = tid >> 4;        // tap index 0..8
        const int l = tid & 15;        // 16 B128 lanes per 64-float row
        const float* g = kern + (size_t)(n * PP + p) * (HH * WW) + h * WW + l * 4;
        // low 32 bits of a generic LDS pointer == LDS byte offset on CDNA5
        const unsigned lds_off =
            (unsigned)(uintptr_t)(&kw_lds[0][0]) + (unsigned)(tid * 16);
        async_b128_to_lds(lds_off, (const void*)g);
    }
    wait_async_all();      // issuing waves: data landed in LDS
    __syncthreads();       // publish to all waves in the workgroup

    // ---- Compute: thread = (one channel, 8 consecutive w) --------------------
    const int c  = c0 + (tid >> 3);    // 32 channels
    const int wq = tid & 7;            // 8 strips of 8 pixels
    const int w0 = wq * 8;

    float acc[8];
#pragma unroll
    for (int i = 0; i < 8; ++i) acc[i] = 0.f;

    const float* xbase = x + (size_t)(n * CC + c) * (HH * WW);

#pragma unroll
    for (int r = 0; r < 3; ++r) {
        const int hr = h + r - 1;                 // block-uniform branch
        if (hr >= 0 && hr < HH) {
            const float* xr = xbase + hr * WW + w0;
            const float4 a = *(const float4*)(xr);
            const float4 b = *(const float4*)(xr + 4);
            float v[10];
            v[0] = (w0 > 0)       ? xr[-1] : 0.f; // left halo / zero-pad
            v[1] = a.x; v[2] = a.y; v[3] = a.z; v[4] = a.w;
            v[5] = b.x; v[6] = b.y; v[7] = b.z; v[8] = b.w;
            v[9] = (w0 + 8 < WW)  ? xr[8]  : 0.f; // right halo / zero-pad
#pragma unroll
            for (int kw = 0; kw < 3; ++kw) {
                const int p = r * 3 + kw;
                const float4 k0 = *(const float4*)&kw_lds[p][w0];
                const float4 k1 = *(const float4*)&kw_lds[p][w0 + 4];
                acc[0] += v[0 + kw] * k0.x;
                acc[1] += v[1 + kw] * k0.y;
                acc[2] += v[2 + kw] * k0.z;
                acc[3] += v[3 + kw] * k0.w;
                acc[4] += v[4 + kw] * k1.x;
                acc[5] += v[5 + kw] * k1.y;
                acc[6] += v[6 + kw] * k1.z;
                acc[7] += v[7 + kw] * k1.w;
            }
        }
    }

    float* orow = out + ((size_t)(n * CC + c) * HH + h) * WW + w0;
    *(float4*)(orow)     = make_float4(acc[0], acc[1], acc[2], acc[3]);
    *(float4*)(orow + 4) = make_float4(acc[4], acc[5], acc[6], acc[7]);
}

extern "C" void kernel_launch(void* const* d_in, const int* in_sizes, int n_in,
                              void* d_out, int out_size, void* d_ws, size_t ws_size,
                              hipStream_t stream) {
    (void)in_sizes; (void)n_in; (void)out_size; (void)d_ws; (void)ws_size;
    const float* x    = (const float*)d_in[0];   // [8,256,64,64] f32
    const float* kern = (const float*)d_in[1];   // [8,1,9,64,64] f32
    float* out        = (float*)d_out;           // [8,256,64,64] f32

    dim3 grid(HH, CC / CB, NN);   // (64, 8, 8) = 4096 blocks
    dim3 block(TPB);
    ddc_kernel<<<grid, block, 0, stream>>>(x, kern, out);
}